// AugmentPipe_10900626997902
// MI455X (gfx1250) — compile-verified
//
#include <hip/hip_runtime.h>

typedef __attribute__((ext_vector_type(2))) float v2f;
typedef __attribute__((ext_vector_type(8))) float v8f;

#define BB 64
#define CC 3
#define HH 256
#define WW 256
#define NBC (BB*CC)
#define SHH 524
#define SWW 524
#define LDSW 544   // 524 data cols + 20 zeroed pad cols (reads reach col 528)

// ---------------- 3x3 matrix helpers ----------------
__device__ __forceinline__ void mm3(const float* A, const float* B, float* C) {
#pragma unroll
  for (int i = 0; i < 3; ++i)
#pragma unroll
    for (int j = 0; j < 3; ++j)
      C[i*3+j] = A[i*3+0]*B[0*3+j] + A[i*3+1]*B[1*3+j] + A[i*3+2]*B[2*3+j];
}
__device__ __forceinline__ void rmul(float* G, const float* M) {
  float T[9]; mm3(G, M, T);
#pragma unroll
  for (int i = 0; i < 9; ++i) G[i] = T[i];
}
__device__ __forceinline__ void lmul(float* G, const float* M) {
  float T[9]; mm3(M, G, T);
#pragma unroll
  for (int i = 0; i < 9; ++i) G[i] = T[i];
}
__device__ __forceinline__ void set_scale(float* M, float sx, float sy) {
  M[0]=sx; M[1]=0.f; M[2]=0.f; M[3]=0.f; M[4]=sy; M[5]=0.f; M[6]=0.f; M[7]=0.f; M[8]=1.f;
}
__device__ __forceinline__ void set_rot(float* M, float th) {
  float c = cosf(th), s = sinf(th);
  M[0]=c; M[1]=-s; M[2]=0.f; M[3]=s; M[4]=c; M[5]=0.f; M[6]=0.f; M[7]=0.f; M[8]=1.f;
}
__device__ __forceinline__ void set_trans(float* M, float tx, float ty) {
  M[0]=1.f; M[1]=0.f; M[2]=tx; M[3]=0.f; M[4]=1.f; M[5]=ty; M[6]=0.f; M[7]=0.f; M[8]=1.f;
}

// ---------------- per-image affine (theta) ----------------
__global__ void theta_kernel(const float* __restrict__ u_flip, const float* __restrict__ u_rot90,
                             const float* __restrict__ u_xint, const float* __restrict__ u_scale,
                             const float* __restrict__ u_rot1, const float* __restrict__ u_aniso,
                             const float* __restrict__ u_rot2, const float* __restrict__ n_xfrac,
                             float* __restrict__ theta) {
  int b = threadIdx.x;
  if (b >= BB) return;
  const float PI = 3.14159265358979323846f;
  float G[9], M[9];
  set_scale(G, floorf(u_flip[b]*2.0f), 1.0f);
  set_rot(M, 0.5f*PI*floorf(u_rot90[b]*4.0f));                         rmul(G, M);
  float tix = (u_xint[2*b+0]*2.0f - 1.0f)*0.125f;
  float tiy = (u_xint[2*b+1]*2.0f - 1.0f)*0.125f;
  set_trans(M, -rintf(tix*(float)WW), -rintf(tiy*(float)HH));          rmul(G, M);
  float s = exp2f(u_scale[b]*0.2f);
  set_scale(M, s, s);                                                  rmul(G, M);
  set_rot(M, (u_rot1[b]*2.0f - 1.0f)*PI);                              rmul(G, M);
  float a = exp2f(u_aniso[b]*0.2f);
  set_scale(M, a, 1.0f/a);                                             rmul(G, M);
  set_rot(M, (u_rot2[b]*2.0f - 1.0f)*PI);                              rmul(G, M);
  set_trans(M, -n_xfrac[2*b+0]*0.125f*(float)WW,
               -n_xfrac[2*b+1]*0.125f*(float)HH);                      rmul(G, M);
  // G = cs(2,2) @ G @ cs(.5,.5)
  set_scale(M, 0.5f, 0.5f);  rmul(G, M);
  set_scale(M, 2.0f, 2.0f);  lmul(G, M);
  // G = ct(-.5,-.5) @ G @ ct(.5,.5)
  set_trans(M, 0.5f, 0.5f);   rmul(G, M);
  set_trans(M, -0.5f, -0.5f); lmul(G, M);
  // G = cs(2/512, 2/512) @ G @ cs(524/2, 524/2)
  set_scale(M, (float)SWW*0.5f, (float)SHH*0.5f);   rmul(G, M);
  set_scale(M, 2.0f/512.0f, 2.0f/512.0f);           lmul(G, M);
#pragma unroll
  for (int i = 0; i < 6; ++i) theta[b*6 + i] = G[i];  // rows 0,1 of G
}

// Banded down-FIR matrix, shared by both down passes, [i][j] layout so the per-chunk
// fragment loads are base + compile-time offset:
//   wmat[i*48 + j] = f[j - 2*i] if 0 <= j-2i < 12 else 0   (i in [0,16), j in [0,48))
// Down-H uses it as B (i = output col); down-V uses it as A (i = output row).
__device__ __forceinline__ void fill_wmat(float* wmat, const float* __restrict__ hz,
                                          int tid, int nthreads) {
  for (int idx = tid; idx < 16*48; idx += nthreads) {
    int i = idx / 48, j = idx - i*48;
    int t = j - 2*i;
    wmat[idx] = ((unsigned)t < 12u) ? hz[t] : 0.0f;
  }
}

// ------- fused: polyphase 2x-up + bilinear grid sample + horizontal down-FIR (WMMA) -------
// Block: (bc, row-tile). Phase 1: 16x524 sampled tile into LDS (each sample is an exact
// 7x7 weighted gather from the source image: bilinear weights folded into the polyphase
// 6-tap up-filter per axis, zero-border validity absorbed into the weights); pad cols
// 524..543 are zero-filled so phase 2 needs no clamping (and never sees garbage LDS).
// Phase 2: 16 waves; wave w computes out[r, 16w..16w+15] = X(16x48) x Wf(48x16) with
// 12 chained V_WMMA_F32_16X16X4_F32 (Wf banded; j>=42 columns all-zero).
__global__ __launch_bounds__(512) void sample_downh_kernel(
    const float* __restrict__ img, const float* __restrict__ theta,
    const float* __restrict__ hz, float* __restrict__ dh) {
  __shared__ float tile[16*LDSW];
  __shared__ float wmat[16*48];
  __shared__ float fs[12];
  __shared__ float th[6];
  int bc = blockIdx.x;
  int rt = blockIdx.y;
  int b  = bc / CC;
  if (threadIdx.x < 12) fs[threadIdx.x] = hz[threadIdx.x];
  if (threadIdx.x >= 32 && threadIdx.x < 38) th[threadIdx.x-32] = theta[b*6 + (threadIdx.x-32)];
  fill_wmat(wmat, hz, threadIdx.x, 512);
  // zero the pad columns so unclamped phase-2 reads are defined (and multiplied by 0 weights)
  for (int idx = threadIdx.x; idx < 16*(LDSW-SWW); idx += 512) {
    int r = idx / (LDSW-SWW), c = idx - r*(LDSW-SWW);
    tile[r*LDSW + SWW + c] = 0.0f;
  }
  __syncthreads();
  const float* ib = img + (size_t)bc*HH*WW;
  float t0 = th[0], t1 = th[1], t2 = th[2], t3 = th[3], t4 = th[4], t5 = th[5];

  for (int idx = threadIdx.x; idx < 16*SWW; idx += 512) {
    int rowl = idx / SWW;
    int ox   = idx - rowl*SWW;
    int oy   = rt*16 + rowl; if (oy > SHH-1) oy = SHH-1;  // dummy rows in last tile
    float xsv = (2.0f*(float)ox + 1.0f)/(float)SWW - 1.0f;
    float ysv = (2.0f*(float)oy + 1.0f)/(float)SHH - 1.0f;
    float gx = t0*xsv + t1*ysv + t2;
    float gy = t3*xsv + t4*ysv + t5;
    float ixf = (gx + 1.0f)*256.0f - 0.5f;
    float iyf = (gy + 1.0f)*256.0f - 0.5f;
    float x0f = floorf(ixf), y0f = floorf(iyf);
    float fx = ixf - x0f,  fy = iyf - y0f;
    int x0 = (int)x0f, y0 = (int)y0f;
    // zero-border validity of the 4 bilinear taps on the (virtual) 512x512 upsampled image
    float wxa = (x0   >= 0 && x0   < 512) ? (1.0f - fx) : 0.0f;
    float wxb = (x0+1 >= 0 && x0+1 < 512) ? fx          : 0.0f;
    float wya = (y0   >= 0 && y0   < 512) ? (1.0f - fy) : 0.0f;
    float wyb = (y0+1 >= 0 && y0+1 < 512) ? fy          : 0.0f;
    int mx = x0 >> 1, px = x0 & 1;   // arithmetic shift: floor div, works for negatives
    int my = y0 >> 1, py = y0 & 1;
    float wh[7], wv[7];
    int cb_, rb_;
    if (px == 0) {                 // x0 even: cols mx-3..mx+3 ; x0 taps f[2u], x0+1 taps f[2u-1]
      cb_ = mx - 3;
#pragma unroll
      for (int u = 0; u < 7; ++u) {
        float w0 = (u < 6)  ? fs[2*u]   : 0.0f;
        float w1 = (u >= 1) ? fs[2*u-1] : 0.0f;
        wh[u] = wxa*w0 + wxb*w1;
      }
    } else {                       // x0 odd: cols mx-2..mx+3
      cb_ = mx - 2;
#pragma unroll
      for (int u = 0; u < 7; ++u)
        wh[u] = (u < 6) ? (wxa*fs[2*u+1] + wxb*fs[2*u]) : 0.0f;
    }
    if (py == 0) {
      rb_ = my - 3;
#pragma unroll
      for (int v = 0; v < 7; ++v) {
        float w0 = (v < 6)  ? fs[2*v]   : 0.0f;
        float w1 = (v >= 1) ? fs[2*v-1] : 0.0f;
        wv[v] = wya*w0 + wyb*w1;
      }
    } else {
      rb_ = my - 2;
#pragma unroll
      for (int v = 0; v < 7; ++v)
        wv[v] = (v < 6) ? (wya*fs[2*v+1] + wyb*fs[2*v]) : 0.0f;
    }
    float acc = 0.0f;
#pragma unroll
    for (int v = 0; v < 7; ++v) {
      int r = rb_ + v;
      float ra = 0.0f;
      if ((unsigned)r < (unsigned)HH) {
        const float* rp = ib + r*WW;
#pragma unroll
        for (int u = 0; u < 7; ++u) {
          int cx = cb_ + u;
          float pv = ((unsigned)cx < (unsigned)WW) ? rp[cx] : 0.0f;
          ra += wh[u]*pv;
        }
      }
      acc += wv[v]*ra;
    }
    tile[rowl*LDSW + ox] = acc;
  }
  __syncthreads();

  // Phase 2: WMMA horizontal down-filter. out[r, ow] = sum_k f[k]*x[r, 2*ow+1+k]
  int lane = threadIdx.x & 31;
  int wv16 = threadIdx.x >> 5;      // wave id 0..15 -> output col tile
  int nloc = lane & 15;
  int half = lane >> 4;
  int arow = lane & 15;             // A-matrix row (same for both lane halves)
  int cb = 32*wv16 + 1;             // first input col of this tile's band
  // base pointers: all per-chunk indices become compile-time immediate offsets
  const float* ta = &tile[arow*LDSW + cb + 2*half];
  const float* wb = &wmat[nloc*48 + 2*half];
  v8f acc8 = {0.f,0.f,0.f,0.f,0.f,0.f,0.f,0.f};
#pragma unroll
  for (int kk = 0; kk < 12; ++kk) {
    v2f Af, Bf;
    Af.x = ta[4*kk];     Af.y = ta[4*kk + 1];
    Bf.x = wb[4*kk];     Bf.y = wb[4*kk + 1];
    acc8 = __builtin_amdgcn_wmma_f32_16x16x4_f32(false, Af, false, Bf, (short)0, acc8,
                                                 false, false);
  }
  float* dbase = dh + (size_t)bc*SHH*256;
#pragma unroll
  for (int v = 0; v < 8; ++v) {
    int row = rt*16 + v + 8*half;   // C/D layout: VGPR v -> M=v (lanes 0-15), M=v+8 (16-31)
    if (row < SHH) dbase[row*256 + wv16*16 + nloc] = acc8[v];
  }
}

// ------- vertical down-FIR (WMMA): out[oy,ow] = sum_k f[k]*dh[2*oy+1+k, ow] -------
// D(16x16) = Wf'(16x48) x X(48x16); A is the banded filter, B the strided dh rows.
__global__ __launch_bounds__(256) void downv_kernel(const float* __restrict__ dh,
                                                    const float* __restrict__ hz,
                                                    float* __restrict__ out) {
  __shared__ float wmat[16*48];
  fill_wmat(wmat, hz, threadIdx.x, 256);
  __syncthreads();
  int lane = threadIdx.x & 31;
  int wid  = threadIdx.x >> 5;
  int tileIdx = blockIdx.x*8 + wid;   // 192*16*16 = 49152 tiles, exact
  int bc  = tileIdx >> 8;
  int rem = tileIdx & 255;
  int ot  = rem >> 4;                 // output row tile
  int wt  = rem & 15;                 // output col tile
  int nloc = lane & 15;
  int half = lane >> 4;
  int r0  = 32*ot + 1;
  int ow0 = wt*16;
  const float* db = dh + (size_t)bc*SHH*256;
  const float* wa = &wmat[nloc*48 + 2*half];   // A[m,j] = f[j-2m], m = lane&15
  v8f acc8 = {0.f,0.f,0.f,0.f,0.f,0.f,0.f,0.f};
#pragma unroll
  for (int kk = 0; kk < 12; ++kk) {
    int jx = 4*kk + 2*half;
    v2f Af, Bf;
    Af.x = wa[4*kk];     Af.y = wa[4*kk + 1];
    // clamp rows: j>=42 has all-zero A column, but never read outside the written buffer
    int rr0 = r0 + jx;     if (rr0 > SHH-1) rr0 = SHH-1;
    int rr1 = r0 + jx + 1; if (rr1 > SHH-1) rr1 = SHH-1;
    Bf.x = db[rr0*256 + ow0 + nloc];
    Bf.y = db[rr1*256 + ow0 + nloc];
    acc8 = __builtin_amdgcn_wmma_f32_16x16x4_f32(false, Af, false, Bf, (short)0, acc8,
                                                 false, false);
  }
  float* ob = out + (size_t)bc*256*256;
#pragma unroll
  for (int v = 0; v < 8; ++v) {
    int row = ot*16 + v + 8*half;
    ob[row*256 + ow0 + nloc] = acc8[v];
  }
}

extern "C" void kernel_launch(void* const* d_in, const int* in_sizes, int n_in,
                              void* d_out, int out_size, void* d_ws, size_t ws_size,
                              hipStream_t stream) {
  (void)in_sizes; (void)n_in; (void)out_size; (void)ws_size;
  const float* images  = (const float*)d_in[0];
  const float* u_flip  = (const float*)d_in[1];
  const float* u_rot90 = (const float*)d_in[2];
  const float* u_xint  = (const float*)d_in[3];
  const float* u_scale = (const float*)d_in[4];
  const float* u_rot1  = (const float*)d_in[5];
  const float* u_aniso = (const float*)d_in[6];
  const float* u_rot2  = (const float*)d_in[7];
  const float* n_xfrac = (const float*)d_in[8];
  const float* hz      = (const float*)d_in[9];

  float* theta = (float*)d_ws;                            // 64*6 floats
  float* dh    = (float*)((char*)d_ws + 4096);            // 192*524*256 floats (~103 MB)
  float* outp  = (float*)d_out;                           // (64,3,256,256) f32

  theta_kernel<<<1, 64, 0, stream>>>(u_flip, u_rot90, u_xint, u_scale,
                                     u_rot1, u_aniso, u_rot2, n_xfrac, theta);

  dim3 gridS(NBC, 33);   // 33 row-tiles of 16 cover 524 rows
  sample_downh_kernel<<<gridS, 512, 0, stream>>>(images, theta, hz, dh);

  downv_kernel<<<6144, 256, 0, stream>>>(dh, hz, outp);
}